// CRF_NN_48095043781147
// MI455X (gfx1250) — compile-verified
//
#include <hip/hip_runtime.h>
#include <hip/hip_bf16.h>
#include <math.h>

#define N_ROWS 8192
#define D_IN   512
#define D_OUT  128
#define N_EDGE 262144
#define N_ITER 5

typedef __attribute__((ext_vector_type(2))) float v2f;
typedef __attribute__((ext_vector_type(8))) float v8f;

// ---------------------------------------------------------------------------
// Kernel 1: x1 = inputs @ emb1, x2 = inputs @ emb2   [8192,512]x[512,128]
// One block per row; thread j accumulates column j for both outputs.
// ---------------------------------------------------------------------------
__global__ void embed_gemm_kernel(const float* __restrict__ inputs,
                                  const float* __restrict__ e1,
                                  const float* __restrict__ e2,
                                  float* __restrict__ x1,
                                  float* __restrict__ x2) {
    const int i = blockIdx.x;
    const int j = threadIdx.x;           // 0..127
    const float* row = inputs + (size_t)i * D_IN;
    float a1 = 0.f, a2 = 0.f;
#pragma unroll 4
    for (int k = 0; k < D_IN; ++k) {
        const float a = row[k];
        a1 = fmaf(a, e1[k * D_OUT + j], a1);
        a2 = fmaf(a, e2[k * D_OUT + j], a2);
    }
    x1[(size_t)i * D_OUT + j] = a1;
    x2[(size_t)i * D_OUT + j] = a2;
}

// ---------------------------------------------------------------------------
// Kernel 2: streaming row stats of softmax(leaky_relu(x1 x2^T)).
// Block = 128 thr = 4 waves; wave w owns rows blockIdx*64 + w*16 .. +15.
// fp32 WMMA 16x16x4; dual accumulators break the WMMA RAW chain; B-fragments
// loaded just-in-time from LDS (compiler merges into ds_load_2addr_b64);
// LDS double-buffered with register prefetch of tile t+1; per-lane online
// softmax (cross-lane merge deferred to after the tile loop).
// __launch_bounds__(128, 1): live set ~135 VGPRs (A frags 64 + acc 16 +
// prefetch 16 + m/s 16 + addressing) — lift the default VGPR budget so the
// allocator doesn't spill two pairs to scratch inside the hot loop.
//
// Fragment indexing (ISA 7.12.2, fp32 16x16x4):
//   A: lane L -> row (L&15), K-offsets (L>>4)*2 + {0,1}
//   B: lane L -> col (L&15), K-offsets (L>>4)*2 + {0,1}   (B[k][n] = x2[n][k])
//   C/D: VGPR v, lane L -> row v + 8*(L>>4), col (L&15)
// ---------------------------------------------------------------------------
__global__ void __launch_bounds__(128, 1)
row_stats_kernel(const float* __restrict__ x1,
                 const float* __restrict__ x2,
                 float* __restrict__ rmax,
                 float* __restrict__ rsum) {
    __shared__ float lds[2][16 * D_OUT];         // 2 x 8 KB x2 tiles

    const int lane    = threadIdx.x & 31;
    const int wave    = threadIdx.x >> 5;
    const int rowbase = blockIdx.x * 64 + wave * 16;
    const int half    = lane >> 4;               // 0 or 1
    const int lsub    = lane & 15;

    // Preload A fragments for full K=128 (32 k-steps of 4)
    v2f a[32];
    const float* arow = x1 + (size_t)(rowbase + lsub) * D_OUT;
#pragma unroll
    for (int kk = 0; kk < 32; ++kk) {
        a[kk] = *(const v2f*)(arow + kk * 4 + half * 2);
    }

    // Per-lane online softmax state: lane sees only column (lane&15) of each
    // tile, so (m,s) partials are private; merged across lanes at the end.
    float m[8], s[8];
#pragma unroll
    for (int v = 0; v < 8; ++v) { m[v] = -__builtin_inff(); s[v] = 0.f; }

    // Stage tile 0
    {
        const float4* src = (const float4*)x2;
        float4* dst = (float4*)lds[0];
#pragma unroll
        for (int q = 0; q < 4; ++q) dst[threadIdx.x + q * 128] = src[threadIdx.x + q * 128];
    }
    __syncthreads();

    const int NT = N_ROWS / 16;                  // 512 tiles
    for (int tile = 0; tile < NT; ++tile) {
        const int cur = tile & 1;

        // Issue global prefetch of tile t+1 into registers (latency hidden
        // under the WMMA burst below); stored to the other LDS buffer after.
        float4 pf0, pf1, pf2, pf3;
        if (tile + 1 < NT) {
            const float4* src = (const float4*)(x2 + (size_t)(tile + 1) * 16 * D_OUT);
            pf0 = src[threadIdx.x];
            pf1 = src[threadIdx.x + 128];
            pf2 = src[threadIdx.x + 256];
            pf3 = src[threadIdx.x + 384];
        }

        // WMMA burst: B fragments loaded just-in-time (no big live array ->
        // no spills); dual accumulators break the D->C RAW chain.
        const float* brow = lds[cur] + lsub * D_OUT + half * 2;
        v8f acc0 = {}, acc1 = {};
#pragma unroll
        for (int kk = 0; kk < 32; kk += 2) {
            const v2f b0 = *(const v2f*)(brow + kk * 4);
            const v2f b1 = *(const v2f*)(brow + kk * 4 + 4);
            acc0 = __builtin_amdgcn_wmma_f32_16x16x4_f32(
                false, a[kk],     false, b0, (short)0, acc0, false, false);
            acc1 = __builtin_amdgcn_wmma_f32_16x16x4_f32(
                false, a[kk + 1], false, b1, (short)0, acc1, false, false);
        }

        // Per-lane leaky_relu + online (m,s) update — no cross-lane traffic.
#pragma unroll
        for (int v = 0; v < 8; ++v) {
            float x = acc0[v] + acc1[v];
            x = x > 0.f ? x : 0.2f * x;
            const float nm = fmaxf(m[v], x);
            s[v] = s[v] * __expf(m[v] - nm) + __expf(x - nm);
            m[v] = nm;
        }

        // Complete the prefetch into the other buffer; one barrier per tile.
        if (tile + 1 < NT) {
            float4* dst = (float4*)lds[1 - cur];
            dst[threadIdx.x]       = pf0;
            dst[threadIdx.x + 128] = pf1;
            dst[threadIdx.x + 256] = pf2;
            dst[threadIdx.x + 384] = pf3;
        }
        __syncthreads();
    }

    // Cross-lane merge of (m,s) partials within each 16-lane half.
#pragma unroll
    for (int v = 0; v < 8; ++v) {
#pragma unroll
        for (int mask = 1; mask <= 8; mask <<= 1) {
            const float om = __shfl_xor(m[v], mask, 32);
            const float os = __shfl_xor(s[v], mask, 32);
            const float nm = fmaxf(m[v], om);
            s[v] = s[v] * __expf(m[v] - nm) + os * __expf(om - nm);
            m[v] = nm;
        }
    }

    if (lsub == 0) {
#pragma unroll
        for (int v = 0; v < 8; ++v) {
            const int r = rowbase + v + half * 8;
            rmax[r] = m[v];
            rsum[r] = s[v];
        }
    }
}

// ---------------------------------------------------------------------------
// Kernel 3: per-edge sim recompute + vals + row-normalize + CSR row counts.
// ---------------------------------------------------------------------------
__global__ void edge_sim_kernel(const float* __restrict__ x1,
                                const float* __restrict__ x2,
                                const float* __restrict__ rmax,
                                const float* __restrict__ rsum,
                                const float* __restrict__ edge_vals,
                                const int* __restrict__ erow,
                                const int* __restrict__ ecol,
                                float* __restrict__ vals,
                                float* __restrict__ normalize,
                                int* __restrict__ rowcnt) {
    const int e = blockIdx.x * blockDim.x + threadIdx.x;
    const int r = erow[e];
    const int c = ecol[e];
    const float4* xr = (const float4*)(x1 + (size_t)r * D_OUT);
    const float4* xc = (const float4*)(x2 + (size_t)c * D_OUT);
    float acc = 0.f;
#pragma unroll
    for (int k = 0; k < D_OUT / 4; ++k) {
        const float4 av = xr[k], bv = xc[k];
        acc = fmaf(av.x, bv.x, acc);
        acc = fmaf(av.y, bv.y, acc);
        acc = fmaf(av.z, bv.z, acc);
        acc = fmaf(av.w, bv.w, acc);
    }
    const float l   = acc > 0.f ? acc : 0.2f * acc;
    const float sim = __expf(l - rmax[r]) / rsum[r];
    const float v   = edge_vals[e] * sim;
    vals[e] = v;
    unsafeAtomicAdd(&normalize[r], v);
    atomicAdd(&rowcnt[r], 1);
}

// ---------------------------------------------------------------------------
// Kernel 4: denom = beta + normalize*alpha + alpha
// ---------------------------------------------------------------------------
__global__ void denom_kernel(const float* __restrict__ normalize,
                             const float* __restrict__ alpha_p,
                             const float* __restrict__ beta_p,
                             float* __restrict__ denomv) {
    const int i = blockIdx.x * blockDim.x + threadIdx.x;
    const float alpha = __expf(alpha_p[0]);
    const float beta  = __expf(beta_p[0]);
    denomv[i] = beta + normalize[i] * alpha + alpha;
}

// ---------------------------------------------------------------------------
// Kernel 5: single-block exclusive scan of rowcnt -> rowstart (N+1) & rowfill.
// 256 threads x 32 rows each.
// ---------------------------------------------------------------------------
__global__ void scan_kernel(const int* __restrict__ rowcnt,
                            int* __restrict__ rowstart,
                            int* __restrict__ rowfill) {
    __shared__ int part[256];
    const int t = threadIdx.x;
    const int base = t * 32;
    int sum = 0;
#pragma unroll
    for (int i = 0; i < 32; ++i) sum += rowcnt[base + i];
    part[t] = sum;
    __syncthreads();
    for (int off = 1; off < 256; off <<= 1) {
        const int add = (t >= off) ? part[t - off] : 0;
        __syncthreads();
        part[t] += add;
        __syncthreads();
    }
    int run = (t > 0) ? part[t - 1] : 0;
#pragma unroll
    for (int i = 0; i < 32; ++i) {
        rowstart[base + i] = run;
        rowfill[base + i]  = run;
        run += rowcnt[base + i];
    }
    if (t == 255) rowstart[N_ROWS] = part[255];
}

// ---------------------------------------------------------------------------
// Kernel 6: CSR fill — permute (col, val) into row-major edge order.
// ---------------------------------------------------------------------------
__global__ void csr_fill_kernel(const int* __restrict__ erow,
                                const int* __restrict__ ecol,
                                const float* __restrict__ vals,
                                int* __restrict__ rowfill,
                                int* __restrict__ col_p,
                                float* __restrict__ vals_p) {
    const int e = blockIdx.x * blockDim.x + threadIdx.x;
    const int r = erow[e];
    const int pos = atomicAdd(&rowfill[r], 1);
    col_p[pos]  = ecol[e];
    vals_p[pos] = vals[e];
}

// ---------------------------------------------------------------------------
// Kernel 7: fused SpMM + update, one block per row, no atomics/memsets.
// out[r,:] = (inputs[r,:]*beta + (sum_e v_e*prev[c_e,:] + prev[r,:])*alpha)/denom[r]
// ---------------------------------------------------------------------------
__global__ void spmm_update_kernel(const int* __restrict__ rowstart,
                                   const int* __restrict__ col_p,
                                   const float* __restrict__ vals_p,
                                   const float* __restrict__ inputs,
                                   const float* __restrict__ prev,
                                   const float* __restrict__ denomv,
                                   const float* __restrict__ alpha_p,
                                   const float* __restrict__ beta_p,
                                   float* __restrict__ outp) {
    const int r = blockIdx.x;
    const int t = threadIdx.x;                    // 0..127, float4 per thread
    const int beg = rowstart[r];
    const int end = rowstart[r + 1];
    float4 acc = make_float4(0.f, 0.f, 0.f, 0.f);
    for (int i = beg; i < end; ++i) {
        const int   c = col_p[i];                 // block-uniform scalar loads
        const float v = vals_p[i];
        const float4 p = ((const float4*)(prev + (size_t)c * D_IN))[t];
        acc.x = fmaf(v, p.x, acc.x);
        acc.y = fmaf(v, p.y, acc.y);
        acc.z = fmaf(v, p.z, acc.z);
        acc.w = fmaf(v, p.w, acc.w);
    }
    const float alpha = __expf(alpha_p[0]);
    const float beta  = __expf(beta_p[0]);
    const float d = denomv[r];
    const float4 pr  = ((const float4*)(prev   + (size_t)r * D_IN))[t];
    const float4 in4 = ((const float4*)(inputs + (size_t)r * D_IN))[t];
    float4 o;
    o.x = (in4.x * beta + (acc.x + pr.x) * alpha) / d;
    o.y = (in4.y * beta + (acc.y + pr.y) * alpha) / d;
    o.z = (in4.z * beta + (acc.z + pr.z) * alpha) / d;
    o.w = (in4.w * beta + (acc.w + pr.w) * alpha) / d;
    ((float4*)(outp + (size_t)r * D_IN))[t] = o;
}

// ---------------------------------------------------------------------------
extern "C" void kernel_launch(void* const* d_in, const int* in_sizes, int n_in,
                              void* d_out, int out_size, void* d_ws, size_t ws_size,
                              hipStream_t stream) {
    const float* inputs    = (const float*)d_in[0];
    const float* emb1      = (const float*)d_in[1];
    const float* emb2      = (const float*)d_in[2];
    const float* alpha_p   = (const float*)d_in[3];
    const float* beta_p    = (const float*)d_in[4];
    const float* edge_vals = (const float*)d_in[5];
    const int*   erow      = (const int*)d_in[6];
    const int*   ecol      = (const int*)d_in[7];

    // Workspace carve (~45 MB)
    float* ws        = (float*)d_ws;
    float* x1        = ws;                                  // N*D_OUT
    float* x2        = x1 + (size_t)N_ROWS * D_OUT;         // N*D_OUT
    float* rmax      = x2 + (size_t)N_ROWS * D_OUT;         // N
    float* rsum      = rmax + N_ROWS;                       // N
    float* vals      = rsum + N_ROWS;                       // E
    float* normalize = vals + N_EDGE;                       // N   (zeroed)
    int*   rowcnt    = (int*)(normalize + N_ROWS);          // N   (zeroed, contiguous)
    float* denomv    = (float*)(rowcnt + N_ROWS);           // N
    int*   rowstart  = (int*)(denomv + N_ROWS);             // N+1
    int*   rowfill   = rowstart + (N_ROWS + 1);             // N
    int*   col_p     = rowfill + N_ROWS;                    // E
    float* vals_p    = (float*)(col_p + N_EDGE);            // E
    float* B0        = vals_p + N_EDGE;                     // N*D_IN
    float* B1        = B0 + (size_t)N_ROWS * D_IN;          // N*D_IN

    embed_gemm_kernel<<<N_ROWS, D_OUT, 0, stream>>>(inputs, emb1, emb2, x1, x2);

    row_stats_kernel<<<N_ROWS / 64, 128, 0, stream>>>(x1, x2, rmax, rsum);

    // zero normalize + rowcnt in one shot (contiguous)
    hipMemsetAsync(normalize, 0, (size_t)N_ROWS * (sizeof(float) + sizeof(int)), stream);

    edge_sim_kernel<<<N_EDGE / 256, 256, 0, stream>>>(x1, x2, rmax, rsum,
                                                      edge_vals, erow, ecol,
                                                      vals, normalize, rowcnt);

    denom_kernel<<<N_ROWS / 256, 256, 0, stream>>>(normalize, alpha_p, beta_p, denomv);

    scan_kernel<<<1, 256, 0, stream>>>(rowcnt, rowstart, rowfill);

    csr_fill_kernel<<<N_EDGE / 256, 256, 0, stream>>>(erow, ecol, vals,
                                                      rowfill, col_p, vals_p);

    const float* prev = inputs;
    for (int it = 0; it < N_ITER; ++it) {
        float* cur = (it == N_ITER - 1) ? (float*)d_out : ((it & 1) ? B1 : B0);
        spmm_update_kernel<<<N_ROWS, 128, 0, stream>>>(rowstart, col_p, vals_p,
                                                       inputs, prev, denomv,
                                                       alpha_p, beta_p, cur);
        prev = cur;
    }
}